// GraphSAGE_36601711296652
// MI455X (gfx1250) — compile-verified
//
#include <hip/hip_runtime.h>
#include <hip/hip_bf16.h>

// ---------------------------------------------------------------------------
// GraphSAGE (2x SAGEConv + BN + ReLU + log_softmax) for MI455X / gfx1250.
// fp32 throughout; GEMM1 uses V_WMMA_F32_16X16X4_F32 (wave32 WMMA).
// M-tile = 48 nodes/block: each B (weight) fragment is reused by 3 WMMA ops,
// cutting L2 weight traffic 3x vs a 16-row tile.
// ---------------------------------------------------------------------------

typedef __attribute__((ext_vector_type(2))) float v2f;
typedef __attribute__((ext_vector_type(8))) float v8f;

#define N_NODES 100000
#define N_EDGES 1600000
#define D 128
#define BN_EPS 1e-5f
#define MTILE 48         // nodes per block in gemm1 (3 x 16-row subtiles)
#define MSUB 3

// ------------------------------ zero ---------------------------------------
__global__ void zero4_kernel(float4* __restrict__ p, int n4) {
    int i = blockIdx.x * blockDim.x + threadIdx.x;
    if (i < n4) p[i] = make_float4(0.f, 0.f, 0.f, 0.f);
}

// ------------------------- edge scatter (mean agg) -------------------------
// One wave32 per edge; each lane moves 4 features (float4 load + 4 f32 atomics
// that resolve in L2 since agg (51MB) is L2-resident).
__global__ __launch_bounds__(256) void scatter_kernel(
    const float* __restrict__ feat, const int* __restrict__ src,
    const int* __restrict__ dst, float* __restrict__ agg,
    float* __restrict__ deg, int n_edges) {
    int wave = (blockIdx.x * blockDim.x + threadIdx.x) >> 5;
    int lane = threadIdx.x & 31;
    if (wave >= n_edges) return;
    int s = src[wave];
    int d = dst[wave];
    float4 v = *(const float4*)(feat + (size_t)s * D + lane * 4);
    float* base = agg + (size_t)d * D + lane * 4;
    atomicAdd(base + 0, v.x);
    atomicAdd(base + 1, v.y);
    atomicAdd(base + 2, v.z);
    atomicAdd(base + 3, v.w);
    if (deg != nullptr && lane == 0) atomicAdd(deg + d, 1.0f);
}

// --------------------- layer-1 GEMM: h1 = [mean|x] @ [W1l;W1r] + b1 --------
// Block = 256 threads (8 waves) handles 48 nodes x 128 outputs.
// A (48x256, mean concat x) staged in LDS with padded rows (stride 260 ->
// conflict-free wmma A-fragment reads). Wave w owns output cols 16w..16w+15
// across all 3 row subtiles; each B fragment feeds 3 WMMAs.
// OOB rows (last block) are zero-filled in LDS so EXEC stays all-ones through
// the WMMA loop; stores/BN stats are guarded per row afterwards.
__global__ __launch_bounds__(256) void gemm1_kernel(
    const float* __restrict__ agg, const float* __restrict__ deg,
    const float* __restrict__ x, const float* __restrict__ W1l,
    const float* __restrict__ W1r, const float* __restrict__ b1,
    float* __restrict__ h1, float* __restrict__ sum1b,
    float* __restrict__ sumsq1b) {
    __shared__ float atile[MTILE][260];   // 48*260*4 = 49,920 B

    const int t = threadIdx.x;
    const int node0 = blockIdx.x * MTILE;

    // Cooperative stage: 48 rows x 256 K (first 128 = agg/deg, next 128 = x)
#pragma unroll
    for (int i = 0; i < (MTILE * 256) / (256 * 4); i++) {  // 12 float4 each
        int f = (i * 256 + t) * 4;  // flat element index
        int m = f >> 8;             // row 0..47
        int k = f & 255;            // K 0..255 (aligned to 4)
        int node = node0 + m;
        float4 v = make_float4(0.f, 0.f, 0.f, 0.f);
        if (node < N_NODES) {
            if (k < 128) {
                v = *(const float4*)(agg + (size_t)node * D + k);
                float invd = 1.0f / fmaxf(deg[node], 1.0f);
                v.x *= invd; v.y *= invd; v.z *= invd; v.w *= invd;
            } else {
                v = *(const float4*)(x + (size_t)node * D + (k - 128));
            }
        }
        atile[m][k + 0] = v.x;
        atile[m][k + 1] = v.y;
        atile[m][k + 2] = v.z;
        atile[m][k + 3] = v.w;
    }
    __syncthreads();

    const int wave = t >> 5;
    const int lane = t & 31;
    const int col0 = wave * 16;
    const int mrow = lane & 15;          // A row / B,N col / C col index
    const int kb = (lane >> 4) * 2;      // K sub-offset per lane half

    v8f acc0 = {}, acc1 = {}, acc2 = {};

    // K = 0..127 : mean part with W1l
#pragma unroll 4
    for (int kk = 0; kk < 128; kk += 4) {
        v2f b;
        const float* Wr = W1l + (size_t)(kk + kb) * D + col0 + mrow;
        b.x = Wr[0];
        b.y = Wr[D];
        v2f a0, a1, a2;
        a0.x = atile[mrow][kk + kb];       a0.y = atile[mrow][kk + kb + 1];
        a1.x = atile[16 + mrow][kk + kb];  a1.y = atile[16 + mrow][kk + kb + 1];
        a2.x = atile[32 + mrow][kk + kb];  a2.y = atile[32 + mrow][kk + kb + 1];
        acc0 = __builtin_amdgcn_wmma_f32_16x16x4_f32(
            false, a0, false, b, (short)0, acc0, false, false);
        acc1 = __builtin_amdgcn_wmma_f32_16x16x4_f32(
            false, a1, false, b, (short)0, acc1, false, false);
        acc2 = __builtin_amdgcn_wmma_f32_16x16x4_f32(
            false, a2, false, b, (short)0, acc2, false, false);
    }
    // K = 128..255 : self part with W1r
#pragma unroll 4
    for (int kk = 0; kk < 128; kk += 4) {
        v2f b;
        const float* Wr = W1r + (size_t)(kk + kb) * D + col0 + mrow;
        b.x = Wr[0];
        b.y = Wr[D];
        v2f a0, a1, a2;
        a0.x = atile[mrow][128 + kk + kb];
        a0.y = atile[mrow][128 + kk + kb + 1];
        a1.x = atile[16 + mrow][128 + kk + kb];
        a1.y = atile[16 + mrow][128 + kk + kb + 1];
        a2.x = atile[32 + mrow][128 + kk + kb];
        a2.y = atile[32 + mrow][128 + kk + kb + 1];
        acc0 = __builtin_amdgcn_wmma_f32_16x16x4_f32(
            false, a0, false, b, (short)0, acc0, false, false);
        acc1 = __builtin_amdgcn_wmma_f32_16x16x4_f32(
            false, a1, false, b, (short)0, acc1, false, false);
        acc2 = __builtin_amdgcn_wmma_f32_16x16x4_f32(
            false, a2, false, b, (short)0, acc2, false, false);
    }

    // Epilogue: bias, store pre-BN h1, binned BN stats (guarded per row)
    const int col = col0 + mrow;
    const float bias = b1[col];
    const int bin = blockIdx.x & 31;

#pragma unroll
    for (int s = 0; s < MSUB; s++) {
        const v8f acc = (s == 0) ? acc0 : (s == 1) ? acc1 : acc2;
        const int rbase = node0 + s * 16 + (lane >> 4) * 8;
        float sm = 0.f, q = 0.f;
#pragma unroll
        for (int v = 0; v < 8; v++) {
            int row = rbase + v;          // C/D layout: VGPR v -> M = v (+8)
            if (row < N_NODES) {
                float val = acc[v] + bias;
                h1[(size_t)row * D + col] = val;
                sm += val;
                q += val * val;
            }
        }
        // lanes l and l^16 hold the same column (different row halves)
        sm += __shfl_xor(sm, 16, 32);
        q += __shfl_xor(q, 16, 32);
        if (lane < 16) {
            atomicAdd(sum1b + bin * D + col, sm);
            atomicAdd(sumsq1b + bin * D + col, q);
        }
    }
}

// ---------------------- BN1: reduce bins -> scale/shift --------------------
__global__ void bn1_finalize_kernel(
    const float* __restrict__ sum1b, const float* __restrict__ sumsq1b,
    const float* __restrict__ gamma1, const float* __restrict__ beta1,
    float* __restrict__ scale1, float* __restrict__ shift1) {
    int c = threadIdx.x;  // 0..127
    float S = 0.f, Q = 0.f;
#pragma unroll
    for (int b = 0; b < 32; b++) {
        S += sum1b[b * D + c];
        Q += sumsq1b[b * D + c];
    }
    const float invN = 1.0f / (float)N_NODES;
    float mu = S * invN;
    float var = Q * invN - mu * mu;
    float sc = gamma1[c] * rsqrtf(var + BN_EPS);
    scale1[c] = sc;
    shift1[c] = beta1[c] - mu * sc;
}

// -------------------------- BN1 apply + ReLU (in place) --------------------
__global__ void bnrelu_kernel(float* __restrict__ h1,
                              const float* __restrict__ scale1,
                              const float* __restrict__ shift1, int n4) {
    int i = blockIdx.x * blockDim.x + threadIdx.x;
    if (i >= n4) return;
    float4 v = *((float4*)h1 + i);
    int c = (i * 4) & (D - 1);
    v.x = fmaxf(0.f, fmaf(v.x, scale1[c + 0], shift1[c + 0]));
    v.y = fmaxf(0.f, fmaf(v.y, scale1[c + 1], shift1[c + 1]));
    v.z = fmaxf(0.f, fmaf(v.z, scale1[c + 2], shift1[c + 2]));
    v.w = fmaxf(0.f, fmaf(v.w, scale1[c + 3], shift1[c + 3]));
    *((float4*)h1 + i) = v;
}

// ---- layer 2: z = mean2 @ W2l + b2 + h1 @ W2r  (N=2, wave per node) -------
// Writes pre-BN logits to d_out and binned column stats for BN2.
__global__ __launch_bounds__(256) void layer2_kernel(
    const float* __restrict__ agg, const float* __restrict__ deg,
    const float* __restrict__ h1, const float* __restrict__ W2l,
    const float* __restrict__ b2, const float* __restrict__ W2r,
    float* __restrict__ out, float* __restrict__ part2, int n_nodes) {
    int node = (blockIdx.x * blockDim.x + threadIdx.x) >> 5;
    int lane = threadIdx.x & 31;
    if (node >= n_nodes) return;

    float invd = 1.0f / fmaxf(deg[node], 1.0f);
    int k = lane * 4;
    float4 m4 = *(const float4*)(agg + (size_t)node * D + k);
    m4.x *= invd; m4.y *= invd; m4.z *= invd; m4.w *= invd;
    float4 h4 = *(const float4*)(h1 + (size_t)node * D + k);

    // W2l/W2r row-major (128,2): rows k..k+3 -> 8 consecutive floats
    const float4 wla = *(const float4*)(W2l + k * 2);
    const float4 wlb = *(const float4*)(W2l + k * 2 + 4);
    const float4 wra = *(const float4*)(W2r + k * 2);
    const float4 wrb = *(const float4*)(W2r + k * 2 + 4);

    float a0 = m4.x * wla.x + m4.y * wla.z + m4.z * wlb.x + m4.w * wlb.z +
               h4.x * wra.x + h4.y * wra.z + h4.z * wrb.x + h4.w * wrb.z;
    float a1 = m4.x * wla.y + m4.y * wla.w + m4.z * wlb.y + m4.w * wlb.w +
               h4.x * wra.y + h4.y * wra.w + h4.z * wrb.y + h4.w * wrb.w;

#pragma unroll
    for (int off = 16; off >= 1; off >>= 1) {
        a0 += __shfl_xor(a0, off, 32);
        a1 += __shfl_xor(a1, off, 32);
    }
    if (lane == 0) {
        float z0 = a0 + b2[0];
        float z1 = a1 + b2[1];
        out[(size_t)node * 2 + 0] = z0;
        out[(size_t)node * 2 + 1] = z1;
        float* bin = part2 + (node & 255) * 4;
        atomicAdd(bin + 0, z0);
        atomicAdd(bin + 1, z1);
        atomicAdd(bin + 2, z0 * z0);
        atomicAdd(bin + 3, z1 * z1);
    }
}

// ---------------------- BN2: reduce 256 bins -> scale/shift ----------------
__global__ __launch_bounds__(256) void finalize2_kernel(
    const float* __restrict__ part2, const float* __restrict__ gamma2,
    const float* __restrict__ beta2, float* __restrict__ scale2,
    float* __restrict__ shift2) {
    __shared__ float red[256][4];
    int t = threadIdx.x;
    float4 v = *(const float4*)(part2 + t * 4);
    red[t][0] = v.x; red[t][1] = v.y; red[t][2] = v.z; red[t][3] = v.w;
    __syncthreads();
    for (int s = 128; s >= 1; s >>= 1) {
        if (t < s) {
#pragma unroll
            for (int j = 0; j < 4; j++) red[t][j] += red[t + s][j];
        }
        __syncthreads();
    }
    if (t == 0) {
        const float invN = 1.0f / (float)N_NODES;
#pragma unroll
        for (int c = 0; c < 2; c++) {
            float mu = red[0][c] * invN;
            float var = red[0][2 + c] * invN - mu * mu;
            float sc = gamma2[c] * rsqrtf(var + BN_EPS);
            scale2[c] = sc;
            shift2[c] = beta2[c] - mu * sc;
        }
    }
}

// ----------------------- BN2 apply + log_softmax (in place) ----------------
__global__ void bn2_logsoftmax_kernel(float* __restrict__ out,
                                      const float* __restrict__ scale2,
                                      const float* __restrict__ shift2,
                                      int n) {
    int i = blockIdx.x * blockDim.x + threadIdx.x;
    if (i >= n) return;
    float z0 = fmaf(out[(size_t)i * 2 + 0], scale2[0], shift2[0]);
    float z1 = fmaf(out[(size_t)i * 2 + 1], scale2[1], shift2[1]);
    float m = fmaxf(z0, z1);
    float lse = m + logf(expf(z0 - m) + expf(z1 - m));
    out[(size_t)i * 2 + 0] = z0 - lse;
    out[(size_t)i * 2 + 1] = z1 - lse;
}

// ---------------------------------------------------------------------------
extern "C" void kernel_launch(void* const* d_in, const int* in_sizes, int n_in,
                              void* d_out, int out_size, void* d_ws,
                              size_t ws_size, hipStream_t stream) {
    const float* x      = (const float*)d_in[0];
    const int*   ei     = (const int*)d_in[1];
    const float* W1l    = (const float*)d_in[2];
    const float* b1     = (const float*)d_in[3];
    const float* W1r    = (const float*)d_in[4];
    const float* gamma1 = (const float*)d_in[5];
    const float* beta1  = (const float*)d_in[6];
    const float* W2l    = (const float*)d_in[7];
    const float* b2     = (const float*)d_in[8];
    const float* W2r    = (const float*)d_in[9];
    const float* gamma2 = (const float*)d_in[10];
    const float* beta2  = (const float*)d_in[11];
    float* out = (float*)d_out;
    float* ws  = (float*)d_ws;

    const int n_edges = in_sizes[1] / 2;
    const int* src = ei;
    const int* dst = ei + n_edges;

    // ---- workspace layout (float offsets) ----
    const size_t AGG     = 0;                         // N*D
    const size_t DEG     = AGG + (size_t)N_NODES * D; // N
    const size_t SUM1B   = DEG + N_NODES;             // 32*128
    const size_t SUMSQ1B = SUM1B + 32 * D;            // 32*128
    const size_t SCALE1  = SUMSQ1B + 32 * D;          // 128
    const size_t SHIFT1  = SCALE1 + D;                // 128
    const size_t PART2   = SHIFT1 + D;                // 256*4
    const size_t SCALE2  = PART2 + 256 * 4;           // 2
    const size_t SHIFT2  = SCALE2 + 2;                // 2
    size_t h1_off = SHIFT2 + 2;
    h1_off = (h1_off + 15) & ~(size_t)15;             // 64B align
    const size_t H1 = h1_off;                         // N*D

    float* agg     = ws + AGG;
    float* deg     = ws + DEG;
    float* sum1b   = ws + SUM1B;
    float* sumsq1b = ws + SUMSQ1B;
    float* scale1  = ws + SCALE1;
    float* shift1  = ws + SHIFT1;
    float* part2   = ws + PART2;
    float* scale2  = ws + SCALE2;
    float* shift2  = ws + SHIFT2;
    float* h1      = ws + H1;

    // 1) zero agg + deg + BN1 bins (contiguous from ws start)
    {
        int n4 = (int)((SUMSQ1B + 32 * D) / 4);  // multiple of 4
        zero4_kernel<<<(n4 + 255) / 256, 256, 0, stream>>>((float4*)ws, n4);
    }
    // 2) edge scatter for layer 1 (also builds degree)
    {
        int blocks = (n_edges + 7) / 8;  // 8 waves of 32 per block
        scatter_kernel<<<blocks, 256, 0, stream>>>(x, src, dst, agg, deg,
                                                   n_edges);
    }
    // 3) WMMA GEMM layer 1 (fused mean, bias, BN stat bins)
    {
        int blocks = (N_NODES + MTILE - 1) / MTILE;
        gemm1_kernel<<<blocks, 256, 0, stream>>>(agg, deg, x, W1l, W1r, b1,
                                                 h1, sum1b, sumsq1b);
    }
    // 4) BN1 finalize
    bn1_finalize_kernel<<<1, D, 0, stream>>>(sum1b, sumsq1b, gamma1, beta1,
                                             scale1, shift1);
    // 5) BN1 + ReLU in place
    {
        int n4 = N_NODES * D / 4;
        bnrelu_kernel<<<(n4 + 255) / 256, 256, 0, stream>>>(h1, scale1, shift1,
                                                            n4);
    }
    // 6) zero agg + layer-2 stat bins
    {
        int n4 = N_NODES * D / 4;
        zero4_kernel<<<(n4 + 255) / 256, 256, 0, stream>>>((float4*)agg, n4);
        zero4_kernel<<<1, 256, 0, stream>>>((float4*)part2, 256);
    }
    // 7) edge scatter for layer 2 (degree already known)
    {
        int blocks = (n_edges + 7) / 8;
        scatter_kernel<<<blocks, 256, 0, stream>>>(h1, src, dst, agg, nullptr,
                                                   n_edges);
    }
    // 8) layer-2 skinny GEMM (wave per node) -> pre-BN logits in d_out
    {
        int blocks = (N_NODES + 7) / 8;
        layer2_kernel<<<blocks, 256, 0, stream>>>(agg, deg, h1, W2l, b2, W2r,
                                                  out, part2, N_NODES);
    }
    // 9) BN2 finalize
    finalize2_kernel<<<1, 256, 0, stream>>>(part2, gamma2, beta2, scale2,
                                            shift2);
    // 10) BN2 + log_softmax in place on d_out
    bn2_logsoftmax_kernel<<<(N_NODES + 255) / 256, 256, 0, stream>>>(
        out, scale2, shift2, N_NODES);
}